// Ranking_Loss_80083960201609
// MI455X (gfx1250) — compile-verified
//
#include <hip/hip_runtime.h>
#include <hip/hip_bf16.h>
#include <math.h>

typedef __attribute__((ext_vector_type(2))) float v2f;
typedef __attribute__((ext_vector_type(8))) float v8f;

#define BB      64
#define HH      384
#define WW      384
#define HWPROD  (HH * WW)
#define EPSV    1e-7f
#define RATIO   1.02f
#define TILE_N  16
#define LSTRIDE 17   // 64 rows x 17 floats: conflict-free LDS layout

__device__ __forceinline__ float softplus_stable(float x) {
    // jax.nn.softplus(x) = max(x,0) + log1p(exp(-|x|))
    return fmaxf(x, 0.0f) + log1pf(expf(-fabsf(x)));
}

// One block handles TILE_N=16 pair indices x all 64 batch images.
// Phase 1: 1024 threads gather + classify, stash masked values in LDS.
// Phase 2: wave 0 reduces over the batch dim with V_WMMA_F32_16X16X4_F32
//          (A = 16x4 f32 data tile, B = ones -> D holds row sums).
__global__ void __launch_bounds__(1024)
pair_loss_kernel(const float* __restrict__ gt, const float* __restrict__ outp,
                 const int* __restrict__ x1, const int* __restrict__ y1,
                 const int* __restrict__ x2, const int* __restrict__ y2,
                 float* __restrict__ ws, int Np)
{
    __shared__ float lterm[BB * LSTRIDE];
    __shared__ float lsq  [BB * LSTRIDE];
    __shared__ float lcnt [BB * LSTRIDE];

    const int tid = threadIdx.x;
    const int b   = tid & 63;    // batch index
    const int m   = tid >> 6;    // pair row within tile (0..15)
    const int n   = blockIdx.x * TILE_N + m;

    float t_term = 0.0f, t_sq = 0.0f, t_cnt = 0.0f;
    if (n < Np) {
        const int i1 = x1[n] * WW + y1[n];
        const int i2 = x2[n] * WW + y2[n];
        const size_t base = (size_t)b * (size_t)HWPROD;
        const float p1 = gt[base + i1];
        const float p2 = gt[base + i2];
        const float o1 = outp[base + i1];
        const float o2 = outp[base + i2];
        float ls = 0.0f;
        if (p1 / (p2 + EPSV) > RATIO)      ls = 1.0f;
        else if (p2 / (p1 + EPSV) > RATIO) ls = -1.0f;
        const float pred = o1 - o2;
        if (ls != 0.0f) {
            t_term = softplus_stable(-ls * pred);
            t_cnt  = 1.0f;
        } else {
            t_sq = pred * pred;
        }
    }
    // transposed layout: [k=b][m], stride 17 -> conflict-free
    lterm[b * LSTRIDE + m] = t_term;
    lsq  [b * LSTRIDE + m] = t_sq;
    lcnt [b * LSTRIDE + m] = t_cnt;
    __syncthreads();

    if (tid < 32) {  // wave-uniform branch: EXEC all-ones inside (WMMA requirement)
        const int lane  = tid;
        const int mrow  = lane & 15;
        const int khalf = lane >> 4;   // lanes 0-15: K0/K1, lanes 16-31: K2/K3
        v8f c_t = {}; v8f c_s = {}; v8f c_c = {};
        const v2f ones = {1.0f, 1.0f};
        #pragma unroll
        for (int s = 0; s < 16; ++s) {   // K chunks of 4 cover b = 0..63
            const int k0 = 4 * s + 2 * khalf;
            v2f a_t = { lterm[k0 * LSTRIDE + mrow], lterm[(k0 + 1) * LSTRIDE + mrow] };
            v2f a_s = { lsq [k0 * LSTRIDE + mrow], lsq [(k0 + 1) * LSTRIDE + mrow] };
            v2f a_c = { lcnt[k0 * LSTRIDE + mrow], lcnt[(k0 + 1) * LSTRIDE + mrow] };
            c_t = __builtin_amdgcn_wmma_f32_16x16x4_f32(false, a_t, false, ones,
                                                        (short)0, c_t, false, false);
            c_s = __builtin_amdgcn_wmma_f32_16x16x4_f32(false, a_s, false, ones,
                                                        (short)0, c_s, false, false);
            c_c = __builtin_amdgcn_wmma_f32_16x16x4_f32(false, a_c, false, ones,
                                                        (short)0, c_c, false, false);
        }
        // D layout: VGPR r of lane L -> M = 8*(L>>4)+r, N = L&15.
        // Column 0 lives in lanes 0 (rows 0-7) and 16 (rows 8-15).
        if ((lane & 15) == 0) {
            const int rbase = khalf * 8;
            #pragma unroll
            for (int r = 0; r < 8; ++r) {
                const float st = c_t[r];
                const float ss = c_s[r];
                const float cn = c_c[r];            // exact integer-valued sum
                const float cz = (float)BB - cn;
                const float log_loss = (cn > 0.0f) ? st / cn : 0.0f;
                const float sq_loss  = (cz > 0.0f) ? ss / cz : 0.0f;
                const int nn = blockIdx.x * TILE_N + rbase + r;
                if (nn < Np) ws[nn] = log_loss + sq_loss;
            }
        }
    }
}

// Deterministic mean over ws[0..Np): fixed strided order + LDS tree.
__global__ void __launch_bounds__(256)
reduce_mean_kernel(const float* __restrict__ ws, float* __restrict__ outv, int Np)
{
    __shared__ float red[256];
    float s = 0.0f;
    for (int i = threadIdx.x; i < Np; i += 256) s += ws[i];
    red[threadIdx.x] = s;
    __syncthreads();
    for (int w = 128; w > 0; w >>= 1) {
        if ((int)threadIdx.x < w) red[threadIdx.x] += red[threadIdx.x + w];
        __syncthreads();
    }
    if (threadIdx.x == 0) outv[0] = red[0] / (float)Np;
}

extern "C" void kernel_launch(void* const* d_in, const int* in_sizes, int n_in,
                              void* d_out, int out_size, void* d_ws, size_t ws_size,
                              hipStream_t stream) {
    const float* gt   = (const float*)d_in[0];
    const float* outp = (const float*)d_in[1];
    const int*   x1   = (const int*)d_in[2];
    const int*   y1   = (const int*)d_in[3];
    const int*   x2   = (const int*)d_in[4];
    const int*   y2   = (const int*)d_in[5];
    const int    Np   = in_sizes[2];

    float* ws = (float*)d_ws;   // Np floats of per-pair loss
    const int nblocks = (Np + TILE_N - 1) / TILE_N;

    pair_loss_kernel<<<nblocks, 1024, 0, stream>>>(gt, outp, x1, y1, x2, y2, ws, Np);
    reduce_mean_kernel<<<1, 256, 0, stream>>>(ws, (float*)d_out, Np);
}